// SeqAttwithPairBias_34660386079011
// MI455X (gfx1250) — compile-verified
//
#include <hip/hip_runtime.h>
#include <hip/hip_bf16.h>
#include <math.h>

// ---- problem constants (match reference) ----
#define SEQ_DIMN 256
#define PAIR_DIMN 128
#define NHEADS 8
#define HID 32
#define BBATCH 4
#define LSEQ 512
#define DMODEL 256            // NHEADS*HID
#define MROWS (BBATCH * LSEQ) // 2048

typedef __attribute__((ext_vector_type(2))) float v2f;
typedef __attribute__((ext_vector_type(8))) float v8f;

// V_WMMA_F32_16X16X4_F32 : D(16x16 f32) = A(16x4 f32) x B(4x16 f32) + C
// 8-arg form: (neg_a, A, neg_b, B, c_mod, C, reuse_a, reuse_b)
__device__ __forceinline__ v8f wmma4(v2f a, v2f b, v8f c) {
  return __builtin_amdgcn_wmma_f32_16x16x4_f32(false, a, false, b, (short)0, c,
                                               false, false);
}

// 8B-aligned pair loads (kh is always even -> p is 8B aligned)
__device__ __forceinline__ v2f ld2(const float* p) {
  return *(const v2f*)p; // global_load_b64
}
__device__ __forceinline__ v2f ld2_nt(const float* p) {
  return __builtin_nontemporal_load((const v2f*)p); // b64, TH=NT (no L2 reuse)
}
__device__ __forceinline__ float fast_rcp(float x) {
  return __builtin_amdgcn_rcpf(x); // v_rcp_f32: plenty for sigmoid/softmax
}

// =====================================================================
// Kernel 1: s = LN(seq); Q = (s@Wq)/sqrt(32); K = s@Wk; V = s@Wv;
//           G = sigmoid(s@Wg + bg).  One 16-row tile per block, 8 waves.
// =====================================================================
__global__ __launch_bounds__(256) void k_ln_qkvg(
    const float* __restrict__ seq, const float* __restrict__ g,
    const float* __restrict__ bta, const float* __restrict__ Wq,
    const float* __restrict__ Wk, const float* __restrict__ Wv,
    const float* __restrict__ Wg, const float* __restrict__ bg,
    float* __restrict__ Q, float* __restrict__ K, float* __restrict__ V,
    float* __restrict__ G) {
  __shared__ float sA[16][260]; // padded: A-layout reads hit distinct banks

  const int tid = threadIdx.x;
  const int lane = tid & 31;
  const int wave = tid >> 5;
  const int hi = lane >> 4;
  const int lo = lane & 15;
  const int i0 = blockIdx.x * 16;

  // ---- LayerNorm: thread (ri, cj) reduces 16 strided columns ----
  const int ri = tid >> 4;
  const int cj = tid & 15;
  float x[16];
  float sum = 0.f, sq = 0.f;
  const float* row = seq + (size_t)(i0 + ri) * SEQ_DIMN;
#pragma unroll
  for (int t = 0; t < 16; ++t) {
    float v = row[cj + 16 * t];
    x[t] = v;
    sum += v;
    sq += v * v;
  }
#pragma unroll
  for (int m = 8; m >= 1; m >>= 1) { // 16 threads of a row share a half-wave
    sum += __shfl_xor(sum, m, 32);
    sq += __shfl_xor(sq, m, 32);
  }
  float mu = sum * (1.f / SEQ_DIMN);
  float var = sq * (1.f / SEQ_DIMN) - mu * mu;
  float rstd = rsqrtf(var + 1e-5f);
#pragma unroll
  for (int t = 0; t < 16; ++t) {
    int c = cj + 16 * t;
    sA[ri][c] = (x[t] - mu) * rstd * g[c] + bta[c];
  }
  __syncthreads();

  const float qscale = 0.17677669529663687f; // 1/sqrt(HID)
#pragma unroll 1
  for (int t = 0; t < 8; ++t) {
    int tile = wave + 8 * t;  // 64 n-tiles: 4 matrices x 16
    int msel = tile >> 4;     // 0..3 -> Q,K,V,G
    int n0 = (tile & 15) * 16;
    const float* W = (msel == 0) ? Wq : (msel == 1) ? Wk : (msel == 2) ? Wv : Wg;
    v8f acc = {};
#pragma unroll 4
    for (int s = 0; s < 64; ++s) {
      int kh = 4 * s + 2 * hi;
      v2f a;
      a.x = sA[lo][kh];
      a.y = sA[lo][kh + 1];
      v2f b;
      b.x = W[(size_t)kh * DMODEL + n0 + lo];
      b.y = W[(size_t)(kh + 1) * DMODEL + n0 + lo];
      acc = wmma4(a, b, acc);
    }
#pragma unroll
    for (int r = 0; r < 8; ++r) {
      int mr = r + 8 * hi; // D layout: row = r + 8*hi, col = n0 + lo
      int col = n0 + lo;
      size_t idx = (size_t)(i0 + mr) * DMODEL + col;
      float v = acc[r];
      if (msel == 0)
        Q[idx] = v * qscale;
      else if (msel == 1)
        K[idx] = v;
      else if (msel == 2)
        V[idx] = v;
      else
        G[idx] = fast_rcp(1.f + __expf(-(v + bg[col])));
    }
  }
}

// =====================================================================
// Kernel 2: fused pair-LN + pair bias + flash attention.
// Block = (i-tile of 16, batch b); 8 waves.
//   Phase 1 (per j-tile): wave w computes bias for pair rows i_loc = w, w+8
//     via WMMA: S[ij,n] = X[ij,p] . (gamma_p * W_b[p,n]); LN folded:
//     bias = rstd*(S - mu*G) + B.  mu/var from the SAME registers the A
//     operand loads (single-pass; shfl_xor(16) merges half-wave stats).
//   Phase 2: wave n = head n; QK^T WMMA with bias tile as C operand,
//     online softmax (butterflies within 16-lane halves == C/D row layout),
//     P -> LDS -> A-layout, AV WMMA into two 16x16 f32 accumulators.
// =====================================================================
__global__ __launch_bounds__(256) void k_attention(
    const float* __restrict__ pair, const float* __restrict__ pg,
    const float* __restrict__ pb, const float* __restrict__ Wb,
    const float* __restrict__ Q, const float* __restrict__ K,
    const float* __restrict__ V, const float* __restrict__ G,
    float* __restrict__ attout) {
  __shared__ float sGWt[16][132];        // (gamma*Wb)^T, cols(n)>=8 zero; padded
  __shared__ float sGn[16], sBn[16];     // sum_p gamma_p*Wb[p,n], sum_p beta_p*Wb[p,n]
  __shared__ float sBias[NHEADS][16][17];
  __shared__ float sP[NHEADS][16][18];   // stride 18: 8B-aligned b64, no conflicts
  __shared__ float sMu[8][2][16], sRs[8][2][16];

  const int tid = threadIdx.x;
  const int lane = tid & 31;
  const int wave = tid >> 5;
  const int hi = lane >> 4;
  const int lo = lane & 15;
  const int bidx = blockIdx.y;
  const int i0 = blockIdx.x * 16;
  const size_t brow = (size_t)bidx * LSEQ;

  // ---- one-time LDS constants ----
  if (tid < 128) {
    float gp = pg[tid];
#pragma unroll
    for (int n = 0; n < 8; ++n) sGWt[n][tid] = gp * Wb[tid * NHEADS + n];
#pragma unroll
    for (int n = 8; n < 16; ++n) sGWt[n][tid] = 0.f;
  } else if (tid < 144) {
    int n = tid - 128;
    float gn = 0.f, bn = 0.f;
    if (n < 8) {
      for (int p = 0; p < PAIR_DIMN; ++p) {
        gn += pg[p] * Wb[p * NHEADS + n];
        bn += pb[p] * Wb[p * NHEADS + n];
      }
    }
    sGn[n] = gn;
    sBn[n] = bn;
  }

  // ---- preload Q tile in A layout for head = wave (Q already *1/sqrt(h)) ----
  v2f qa[8];
#pragma unroll
  for (int s = 0; s < 8; ++s) {
    qa[s] = ld2(Q + (brow + i0 + lo) * DMODEL + wave * HID + 4 * s + 2 * hi);
  }

  float mrow[8], lrow[8];
  v8f o0 = {}, o1 = {};
#pragma unroll
  for (int r = 0; r < 8; ++r) {
    mrow[r] = -1e30f;
    lrow[r] = 0.f;
  }

  __syncthreads();

  for (int jt = 0; jt < LSEQ / 16; ++jt) {
    const int j0 = jt * 16;

    // ---------- Phase 1: pair LN + bias GEMM (all heads) ----------
#pragma unroll 1
    for (int u = 0; u < 2; ++u) {
      const int il = wave + 8 * u; // i_loc handled by this wave
      const float* xrow =
          pair + ((brow + i0 + il) * LSEQ + (j0 + lo)) * PAIR_DIMN;
      if (jt + 1 < LSEQ / 16) // prefetch next j-tile row (streamed once)
        __builtin_prefetch(xrow + 16 * PAIR_DIMN, 0, 1);
      v8f s8 = {};
      float psum = 0.f, psq = 0.f;
#pragma unroll 8
      for (int s = 0; s < 32; ++s) {
        int kh = 4 * s + 2 * hi;
        v2f a = ld2_nt(xrow + kh); // single b64 NT load on the hot stream
        v2f b;
        b.x = sGWt[lo][kh];
        b.y = sGWt[lo][kh + 1];
        psum += a.x + a.y;
        psq += a.x * a.x + a.y * a.y;
        s8 = wmma4(a, b, s8);
      }
      // each lane saw 64 of 128 channels of row j_loc=lo; partner has the rest
      psum += __shfl_xor(psum, 16, 32);
      psq += __shfl_xor(psq, 16, 32);
      float mu = psum * (1.f / PAIR_DIMN);
      float var = psq * (1.f / PAIR_DIMN) - mu * mu;
      float rstd = rsqrtf(var + 1e-5f);
      if (hi == 0) {
        sMu[wave][u][lo] = mu;   // stats indexed by j_loc; redistribute for
        sRs[wave][u][lo] = rstd; // the D-layout rows below (same wave, in-order)
      }
      float Gn = sGn[lo]; // head n = lo (cols >= 8 are the zero padding)
      float Bn = sBn[lo];
#pragma unroll
      for (int r = 0; r < 8; ++r) {
        int jl = r + 8 * hi; // D layout: M = j_loc, N = head
        float bias = sRs[wave][u][jl] * (s8[r] - sMu[wave][u][jl] * Gn) + Bn;
        if (lo < 8) sBias[lo][il][jl] = bias;
      }
    }
    __syncthreads(); // bias tile ready for all heads

    // ---------- Phase 2: head = wave ----------
    v8f lg;
#pragma unroll
    for (int r = 0; r < 8; ++r) lg[r] = sBias[wave][r + 8 * hi][lo]; // C = bias
#pragma unroll
    for (int s = 0; s < 8; ++s) { // logits = Qs.K^T + bias
      v2f b = ld2(K + (brow + j0 + lo) * DMODEL + wave * HID + 4 * s + 2 * hi);
      lg = wmma4(qa[s], b, lg);
    }
    // online softmax; rows of the D tile live across 16-lane halves
#pragma unroll
    for (int r = 0; r < 8; ++r) {
      float v = lg[r];
      float mx = v;
      mx = fmaxf(mx, __shfl_xor(mx, 1, 32));
      mx = fmaxf(mx, __shfl_xor(mx, 2, 32));
      mx = fmaxf(mx, __shfl_xor(mx, 4, 32));
      mx = fmaxf(mx, __shfl_xor(mx, 8, 32));
      float mnew = fmaxf(mrow[r], mx);
      float scale = __expf(mrow[r] - mnew);
      float p = __expf(v - mnew);
      float rs = p;
      rs += __shfl_xor(rs, 1, 32);
      rs += __shfl_xor(rs, 2, 32);
      rs += __shfl_xor(rs, 4, 32);
      rs += __shfl_xor(rs, 8, 32);
      lrow[r] = lrow[r] * scale + rs;
      mrow[r] = mnew;
      o0[r] = o0[r] * scale;
      o1[r] = o1[r] * scale;
      sP[wave][r + 8 * hi][lo] = p; // wave-private slab: no barrier needed
    }
    // O += P @ V  (P re-read in A layout; V fetched in B layout)
#pragma unroll
    for (int s = 0; s < 4; ++s) {
      int kh = 4 * s + 2 * hi;
      v2f a;
      a.x = sP[wave][lo][kh];
      a.y = sP[wave][lo][kh + 1];
      size_t v0 = (brow + j0 + kh) * DMODEL + wave * HID + lo;
      v2f b0, b1;
      b0.x = V[v0];
      b0.y = V[v0 + DMODEL];
      b1.x = V[v0 + 16];
      b1.y = V[v0 + DMODEL + 16];
      o0 = wmma4(a, b0, o0);
      o1 = wmma4(a, b1, o1);
    }
    __syncthreads(); // protect sBias before next iteration overwrites it
  }

  // ---- epilogue: normalize, gate, store ----
#pragma unroll
  for (int r = 0; r < 8; ++r) {
    int mr = r + 8 * hi;
    float inv = fast_rcp(lrow[r]);
    size_t base = (brow + i0 + mr) * DMODEL + wave * HID + lo;
    float g0 = G[base];
    float g1 = G[base + 16];
    attout[base] = o0[r] * inv * g0;
    attout[base + 16] = o1[r] * inv * g1;
  }
}

// =====================================================================
// Kernel 3: out = attout @ W_o + b_o   (2048x256 @ 256x256)
// =====================================================================
__global__ __launch_bounds__(256) void k_outproj(const float* __restrict__ X,
                                                 const float* __restrict__ Wo,
                                                 const float* __restrict__ bo,
                                                 float* __restrict__ out) {
  const int tid = threadIdx.x;
  const int lane = tid & 31;
  const int wave = tid >> 5;
  const int hi = lane >> 4;
  const int lo = lane & 15;
  const int i0 = blockIdx.x * 16;

#pragma unroll 1
  for (int t = 0; t < 2; ++t) {
    int n0 = (wave * 2 + t) * 16;
    v8f acc = {};
#pragma unroll 4
    for (int s = 0; s < 64; ++s) {
      int kh = 4 * s + 2 * hi;
      v2f a = ld2(X + (size_t)(i0 + lo) * DMODEL + kh);
      v2f b;
      b.x = Wo[(size_t)kh * SEQ_DIMN + n0 + lo];
      b.y = Wo[(size_t)(kh + 1) * SEQ_DIMN + n0 + lo];
      acc = wmma4(a, b, acc);
    }
#pragma unroll
    for (int r = 0; r < 8; ++r) {
      int mr = r + 8 * hi;
      int col = n0 + lo;
      out[(size_t)(i0 + mr) * SEQ_DIMN + col] = acc[r] + bo[col];
    }
  }
}

// =====================================================================
extern "C" void kernel_launch(void* const* d_in, const int* in_sizes, int n_in,
                              void* d_out, int out_size, void* d_ws,
                              size_t ws_size, hipStream_t stream) {
  const float* seq = (const float*)d_in[0];
  const float* pair = (const float*)d_in[1];
  const float* ln_seq_g = (const float*)d_in[2];
  const float* ln_seq_b = (const float*)d_in[3];
  const float* ln_pair_g = (const float*)d_in[4];
  const float* ln_pair_b = (const float*)d_in[5];
  const float* W_b = (const float*)d_in[6];
  const float* W_q = (const float*)d_in[7];
  const float* W_k = (const float*)d_in[8];
  const float* W_v = (const float*)d_in[9];
  const float* W_g = (const float*)d_in[10];
  const float* b_g = (const float*)d_in[11];
  const float* W_o = (const float*)d_in[12];
  const float* b_o = (const float*)d_in[13];

  float* ws = (float*)d_ws;
  const size_t NM = (size_t)MROWS * DMODEL; // 2048*256 floats per buffer
  float* Q = ws;
  float* K = ws + NM;
  float* V = ws + 2 * NM;
  float* G = ws + 3 * NM;
  float* ATT = ws + 4 * NM; // total ws use: 5*2MB = 10MB

  k_ln_qkvg<<<dim3(MROWS / 16), dim3(256), 0, stream>>>(
      seq, ln_seq_g, ln_seq_b, W_q, W_k, W_v, W_g, b_g, Q, K, V, G);
  k_attention<<<dim3(LSEQ / 16, BBATCH), dim3(256), 0, stream>>>(
      pair, ln_pair_g, ln_pair_b, W_b, Q, K, V, G, ATT);
  k_outproj<<<dim3(MROWS / 16), dim3(256), 0, stream>>>(ATT, W_o, b_o,
                                                        (float*)d_out);
}